// DecoderLayer_60567628808537
// MI455X (gfx1250) — compile-verified
//
#include <hip/hip_runtime.h>
#include <math.h>

typedef __attribute__((ext_vector_type(2))) float v2f;
typedef __attribute__((ext_vector_type(8))) float v8f;

namespace {
constexpr int B_N = 4;      // batch
constexpr int T_L = 2048;   // target length
constexpr int T_S = 4096;   // memory length
constexpr int CH  = 256;    // channels
constexpr int NHD = 8;      // heads
constexpr int HD  = 32;     // head dim
constexpr int MLP = 1024;   // mlp hidden
constexpr int NL  = B_N * T_L;   // 8192 rows
constexpr int NS  = B_N * T_S;   // 16384 rows
constexpr float LN_EPS = 1e-5f;
constexpr float LA_EPS = 1e-6f;
constexpr float SM_SCALE = 0.17677669529663687f; // 32^-0.5
}

// ---------------- WMMA helper (fp32 16x16x4, full-precision accumulate) -----
__device__ __forceinline__ v8f wmma4(v2f a, v2f b, v8f c) {
  // (neg_a, A, neg_b, B, c_mod, C, reuse_a, reuse_b)
  return __builtin_amdgcn_wmma_f32_16x16x4_f32(false, a, false, b, (short)0, c,
                                               false, false);
}

__device__ __forceinline__ float rmax16(float v) {
  v = fmaxf(v, __shfl_xor(v, 1, 32));
  v = fmaxf(v, __shfl_xor(v, 2, 32));
  v = fmaxf(v, __shfl_xor(v, 4, 32));
  v = fmaxf(v, __shfl_xor(v, 8, 32));
  return v;
}
__device__ __forceinline__ float rsum16(float v) {
  v += __shfl_xor(v, 1, 32);
  v += __shfl_xor(v, 2, 32);
  v += __shfl_xor(v, 4, 32);
  v += __shfl_xor(v, 8, 32);
  return v;
}
__device__ __forceinline__ float elup1(float x) {   // elu(x)+1
  return x > 0.0f ? x + 1.0f : __expf(x);
}

// ---------------- generic WMMA GEMM: out = [res +] A@B [+ bias] -------------
// A: [M,K] row-major, B: [K,Nout] row-major. One wave -> 32x64 tile,
// software-pipelined: next k-step fragments loaded before current WMMAs issue.
__global__ void gemm_wmma(const float* __restrict__ A, const float* __restrict__ B,
                          const float* __restrict__ bias, const float* __restrict__ res,
                          float* __restrict__ out, int M, int K, int Nout) {
  const int lane = threadIdx.x & 31;
  const int half = lane >> 4;
  const int li   = lane & 15;
  const int m0   = blockIdx.x * 32;
  const size_t arow0 = (size_t)(m0 + li) * (size_t)K;
  const size_t arow1 = arow0 + (size_t)16 * K;
  const int colbase = blockIdx.y * 64 + li;

  v8f acc[8];
  #pragma unroll
  for (int i = 0; i < 8; ++i) acc[i] = (v8f){0, 0, 0, 0, 0, 0, 0, 0};

  // prologue: fragments for k-step 0
  v2f a0c, a1c, bc[4];
  {
    const int ka = 2 * half;
    a0c = *(const v2f*)(A + arow0 + ka);
    a1c = *(const v2f*)(A + arow1 + ka);
    const float* b0p = B + (size_t)ka * Nout + colbase;
    const float* b1p = b0p + Nout;
    #pragma unroll
    for (int j = 0; j < 4; ++j) { bc[j].x = b0p[j * 16]; bc[j].y = b1p[j * 16]; }
  }
  for (int k0 = 0; k0 + 4 < K; k0 += 4) {
    // issue next k-step loads first so they overlap the 8 WMMAs below
    const int ka = k0 + 4 + 2 * half;
    v2f a0n = *(const v2f*)(A + arow0 + ka);
    v2f a1n = *(const v2f*)(A + arow1 + ka);
    v2f bn[4];
    const float* b0p = B + (size_t)ka * Nout + colbase;
    const float* b1p = b0p + Nout;
    #pragma unroll
    for (int j = 0; j < 4; ++j) { bn[j].x = b0p[j * 16]; bn[j].y = b1p[j * 16]; }
    __builtin_prefetch(A + arow0 + k0 + 64, 0, 1);   // global_prefetch_b8
    #pragma unroll
    for (int j = 0; j < 4; ++j) acc[j]     = wmma4(a0c, bc[j], acc[j]);
    #pragma unroll
    for (int j = 0; j < 4; ++j) acc[4 + j] = wmma4(a1c, bc[j], acc[4 + j]);
    a0c = a0n; a1c = a1n;
    #pragma unroll
    for (int j = 0; j < 4; ++j) bc[j] = bn[j];
  }
  #pragma unroll
  for (int j = 0; j < 4; ++j) acc[j]     = wmma4(a0c, bc[j], acc[j]);
  #pragma unroll
  for (int j = 0; j < 4; ++j) acc[4 + j] = wmma4(a1c, bc[j], acc[4 + j]);

  const bool hb = (bias != nullptr), hr = (res != nullptr);
  float bi[4];
  #pragma unroll
  for (int j = 0; j < 4; ++j) bi[j] = hb ? bias[colbase + j * 16] : 0.0f;
  #pragma unroll
  for (int rb = 0; rb < 2; ++rb) {
    #pragma unroll
    for (int r = 0; r < 8; ++r) {
      const int m = m0 + rb * 16 + r + 8 * half;
      const size_t o = (size_t)m * Nout + colbase;
      #pragma unroll
      for (int j = 0; j < 4; ++j) {
        float v = acc[rb * 4 + j][r] + bi[j];
        if (hr) v += res[o + j * 16];
        out[o + j * 16] = v;
      }
    }
  }
}

// ---------------- LayerNorm (one 256-thread block per row) ------------------
__global__ void ln_kernel(const float* __restrict__ x, const float* __restrict__ g,
                          const float* __restrict__ b, const float* __restrict__ pos,
                          float* __restrict__ y, float* __restrict__ y2) {
  __shared__ float red[CH];
  const int t = threadIdx.x;
  const size_t base = (size_t)blockIdx.x * CH;
  float xv = x[base + t];
  red[t] = xv; __syncthreads();
  for (int s = CH / 2; s > 0; s >>= 1) { if (t < s) red[t] += red[t + s]; __syncthreads(); }
  const float mean = red[0] * (1.0f / CH); __syncthreads();
  const float d = xv - mean;
  red[t] = d * d; __syncthreads();
  for (int s = CH / 2; s > 0; s >>= 1) { if (t < s) red[t] += red[t + s]; __syncthreads(); }
  const float var = red[0] * (1.0f / CH);
  const float yv = d * rsqrtf(var + LN_EPS) * g[t] + b[t];
  y[base + t] = yv;
  if (y2) y2[base + t] = yv + pos[base + t];
}

// ---------------- mem = memory + pos_embed (pos broadcast over batch) -------
__global__ void add_pos(const float* __restrict__ a, const float* __restrict__ p,
                        float* __restrict__ o) {
  const size_t total = (size_t)NS * CH, pm = (size_t)T_S * CH;
  for (size_t i = (size_t)blockIdx.x * blockDim.x + threadIdx.x; i < total;
       i += (size_t)gridDim.x * blockDim.x)
    o[i] = a[i] + p[i % pm];
}

// ---------------- linear-attn KV + Ksum reduction (tiny op) -----------------
// kv[n,h,d,e] = sum_l elup1(k[n,l,h,d]) * v[n,l,h,e]; ksum[n,h,d] = sum_l elup1(k)
__global__ void kv_kernel(const float* __restrict__ kbuf, const float* __restrict__ vbuf,
                          float* __restrict__ kv, float* __restrict__ ksum) {
  const int nh = blockIdx.x, n = nh / NHD, h = nh % NHD;
  const int t = threadIdx.x, d = t >> 3, e0 = (t & 7) * 4;
  const float* kb = kbuf + (size_t)n * T_L * CH + h * HD + d;
  const float* vb = vbuf + (size_t)n * T_L * CH + h * HD + e0;
  float a0 = 0, a1 = 0, a2 = 0, a3 = 0, ks = 0;
  for (int l = 0; l < T_L; ++l) {
    const float kf = elup1(kb[(size_t)l * CH]);
    const float4 vv = *(const float4*)(vb + (size_t)l * CH);
    a0 += kf * vv.x; a1 += kf * vv.y; a2 += kf * vv.z; a3 += kf * vv.w;
    ks += kf;
  }
  float* kvp = kv + (size_t)nh * HD * HD + d * HD + e0;
  kvp[0] = a0; kvp[1] = a1; kvp[2] = a2; kvp[3] = a3;
  if ((t & 7) == 0) ksum[nh * HD + d] = ks;
}

// ---------------- linear-attn message: msg = elup1(Q) @ KV * Z (WMMA) -------
__global__ void la_msg(const float* __restrict__ qbuf, const float* __restrict__ kv,
                       const float* __restrict__ ksum, float* __restrict__ msg) {
  const int lane = threadIdx.x & 31, half = lane >> 4, li = lane & 15;
  const int nh = blockIdx.y, n = nh / NHD, h = nh % NHD;
  const int m0 = blockIdx.x * 16;
  const float* Qrow = qbuf + ((size_t)n * T_L + m0 + li) * CH + h * HD;
  const float* KVb = kv + (size_t)nh * HD * HD;
  const float* KS  = ksum + nh * HD;

  // preload all fragments, then chain WMMAs
  v2f kvf0[8], kvf1[8], qf[8];
  #pragma unroll
  for (int kk = 0; kk < 8; ++kk) {
    const int ka = 4 * kk + 2 * half;
    const float* b0p = KVb + ka * HD;
    const float* b1p = b0p + HD;
    kvf0[kk].x = b0p[li];      kvf0[kk].y = b1p[li];
    kvf1[kk].x = b0p[16 + li]; kvf1[kk].y = b1p[16 + li];
    v2f qr = *(const v2f*)(Qrow + ka);
    qf[kk].x = elup1(qr.x); qf[kk].y = elup1(qr.y);
  }
  float denom = 0.0f;                    // row li (duplicated across halves)
  #pragma unroll
  for (int d = 0; d < HD; ++d) denom += elup1(Qrow[d]) * KS[d];

  v8f o0 = {0, 0, 0, 0, 0, 0, 0, 0}, o1 = o0;
  #pragma unroll
  for (int kk = 0; kk < 8; ++kk) o0 = wmma4(qf[kk], kvf0[kk], o0);
  #pragma unroll
  for (int kk = 0; kk < 8; ++kk) o1 = wmma4(qf[kk], kvf1[kk], o1);

  #pragma unroll
  for (int r = 0; r < 8; ++r) {
    const int row = r + 8 * half;
    const float z = 1.0f / (__shfl(denom, row, 32) + LA_EPS);
    const size_t o = ((size_t)n * T_L + m0 + row) * CH + h * HD + li;
    msg[o]      = o0[r] * z;
    msg[o + 16] = o1[r] * z;
  }
}

// ---------------- flash cross-attention (softmax MHA), per-wave 16 rows -----
__global__ void flash_attn(const float* __restrict__ Qb, const float* __restrict__ Kb,
                           const float* __restrict__ Vb, float* __restrict__ Ob) {
  const int lane = threadIdx.x & 31, half = lane >> 4, li = lane & 15;
  const int nh = blockIdx.y, n = nh / NHD, h = nh % NHD;
  const int m0 = blockIdx.x * 16;
  const float* Qrow  = Qb + ((size_t)n * T_L + m0 + li) * CH + h * HD;
  const float* Kbase = Kb + (size_t)n * T_S * CH + h * HD;
  const float* Vbase = Vb + (size_t)n * T_S * CH + h * HD;

  // Q fragments are invariant over the whole S loop: load once.
  v2f qf[8];
  #pragma unroll
  for (int kk = 0; kk < 8; ++kk) qf[kk] = *(const v2f*)(Qrow + 4 * kk + 2 * half);

  __shared__ float pl[16 * 16];
  float mi[8], lsum[8];
  #pragma unroll
  for (int r = 0; r < 8; ++r) { mi[r] = -INFINITY; lsum[r] = 0.0f; }
  v8f o0 = {0, 0, 0, 0, 0, 0, 0, 0}, o1 = o0;

  for (int s0 = 0; s0 < T_S; s0 += 16) {
    // ---- P = Q @ K^T: batch-load 8 K fragments, then chain 8 WMMAs ----
    const float* Krow = Kbase + (size_t)(s0 + li) * CH;
    v2f kf[8];
    #pragma unroll
    for (int kk = 0; kk < 8; ++kk) kf[kk] = *(const v2f*)(Krow + 4 * kk + 2 * half);
    v8f p = {0, 0, 0, 0, 0, 0, 0, 0};
    #pragma unroll
    for (int kk = 0; kk < 8; ++kk) p = wmma4(qf[kk], kf[kk], p);

    // ---- preload V fragments (independent of softmax -> latency hidden) ----
    v2f vf0[4], vf1[4];
    #pragma unroll
    for (int t = 0; t < 4; ++t) {
      const int ka = 4 * t + 2 * half;
      const float* v0p = Vbase + (size_t)(s0 + ka) * CH;
      const float* v1p = v0p + CH;
      vf0[t].x = v0p[li];      vf0[t].y = v1p[li];
      vf1[t].x = v0p[16 + li]; vf1[t].y = v1p[16 + li];
    }
    __builtin_prefetch(Krow + 16 * CH, 0, 1);   // warm next K tile

    // ---- online softmax over this 16-col slab ----
    float alpha[8];
    #pragma unroll
    for (int r = 0; r < 8; ++r) {
      const float sc = p[r] * SM_SCALE;
      const float mn = fmaxf(mi[r], rmax16(sc));
      const float e  = __expf(sc - mn);
      const float rs = rsum16(e);
      const float al = __expf(mi[r] - mn);
      lsum[r] = lsum[r] * al + rs;
      mi[r] = mn; alpha[r] = al;
      pl[(r + 8 * half) * 16 + li] = e;   // stage P for A-fragment reload
    }
    #pragma unroll
    for (int r = 0; r < 8; ++r) { o0[r] *= alpha[r]; o1[r] *= alpha[r]; }
    __syncthreads();

    // ---- O += P @ V ----
    v2f af[4];
    #pragma unroll
    for (int t = 0; t < 4; ++t) af[t] = *(const v2f*)(pl + li * 16 + 4 * t + 2 * half);
    #pragma unroll
    for (int t = 0; t < 4; ++t) o0 = wmma4(af[t], vf0[t], o0);
    #pragma unroll
    for (int t = 0; t < 4; ++t) o1 = wmma4(af[t], vf1[t], o1);
    __syncthreads();
  }
  #pragma unroll
  for (int r = 0; r < 8; ++r) {
    const int m = m0 + r + 8 * half;
    const float inv = 1.0f / lsum[r];
    const size_t o = ((size_t)n * T_L + m) * CH + h * HD + li;
    Ob[o]      = o0[r] * inv;
    Ob[o + 16] = o1[r] * inv;
  }
}

// ---------------- depthwise 3-tap conv over points + bias + exact GELU ------
__global__ void dwconv_gelu(const float* __restrict__ hin, const float* __restrict__ dwk,
                            const float* __restrict__ dwb, float* __restrict__ out) {
  const size_t total = (size_t)NL * MLP;
  for (size_t i = (size_t)blockIdx.x * blockDim.x + threadIdx.x; i < total;
       i += (size_t)gridDim.x * blockDim.x) {
    const int f = (int)(i % MLP);
    const int p = (int)((i / MLP) % T_L);
    // 3x3 kernel over (points,1) grid with pad 1: only center column survives
    const float w0 = dwk[f * 9 + 1], w1 = dwk[f * 9 + 4], w2 = dwk[f * 9 + 7];
    float acc = hin[i] * w1 + dwb[f];
    if (p > 0)        acc += hin[i - MLP] * w0;
    if (p < T_L - 1)  acc += hin[i + MLP] * w2;
    out[i] = acc * 0.5f * (1.0f + erff(acc * 0.70710678118654752f));
  }
}

// ---------------- host-side orchestration -----------------------------------
extern "C" void kernel_launch(void* const* d_in, const int* in_sizes, int n_in,
                              void* d_out, int out_size, void* d_ws, size_t ws_size,
                              hipStream_t stream) {
  (void)in_sizes; (void)n_in; (void)out_size; (void)ws_size;
  const float* tgt       = (const float*)d_in[0];
  const float* memory    = (const float*)d_in[1];
  const float* tgt_pos   = (const float*)d_in[2];
  const float* pos_embed = (const float*)d_in[3];
  const float* ln1_g = (const float*)d_in[4],  *ln1_b = (const float*)d_in[5];
  const float* ln2_g = (const float*)d_in[6],  *ln2_b = (const float*)d_in[7];
  const float* ln3_g = (const float*)d_in[8],  *ln3_b = (const float*)d_in[9];
  const float* wq = (const float*)d_in[10], *bq = (const float*)d_in[11];
  const float* wk = (const float*)d_in[12], *bk = (const float*)d_in[13];
  const float* wv = (const float*)d_in[14], *bv = (const float*)d_in[15];
  const float* w_merge = (const float*)d_in[16];
  const float* cwq = (const float*)d_in[17], *cbq = (const float*)d_in[18];
  const float* cwk = (const float*)d_in[19], *cbk = (const float*)d_in[20];
  const float* cwv = (const float*)d_in[21], *cbv = (const float*)d_in[22];
  const float* cwo = (const float*)d_in[23], *cbo = (const float*)d_in[24];
  const float* mw1 = (const float*)d_in[25], *mb1 = (const float*)d_in[26];
  const float* dw_k = (const float*)d_in[27], *dw_b = (const float*)d_in[28];
  const float* mw2 = (const float*)d_in[29], *mb2 = (const float*)d_in[30];
  float* outp = (float*)d_out;

  const size_t NLC = (size_t)NL * CH;   // 2M floats
  const size_t NSC = (size_t)NS * CH;   // 4M floats
  float* W    = (float*)d_ws;
  float* res1 = W;               // after self-attn residual
  float* res2 = res1 + NLC;      // after cross-attn residual
  float* ta   = res2 + NLC;      // tgt2 / x
  float* tb   = ta + NLC;        // qk / msg ; stage C: h2 spans tb..te (8M)
  float* tc   = tb + NLC;        // q / cq
  float* td   = tc + NLC;        // k / attn_out
  float* te   = td + NLC;        // v
  float* bg0  = te + NLC;        // mem ; stage C: h (spans 2 NSC)
  float* bg1  = bg0 + NSC;       // ck
  float* bg2  = bg1 + NSC;       // cv
  float* kvb  = bg2 + NSC;       // [N,H,32,32]
  float* ksb  = kvb + (size_t)B_N * NHD * HD * HD; // [N,H,32]

  auto gemm = [&](const float* A, const float* Bm, const float* bias,
                  const float* res, float* out, int M, int K, int Nout) {
    dim3 g(M / 32, Nout / 64);
    gemm_wmma<<<g, 32, 0, stream>>>(A, Bm, bias, res, out, M, K, Nout);
  };

  // ---- self attention (linear attention) ----
  ln_kernel<<<NL, CH, 0, stream>>>(tgt, ln1_g, ln1_b, tgt_pos, ta, tb); // ta=tgt2, tb=qk
  gemm(tb, wq, bq, nullptr, tc, NL, CH, CH);                            // q
  gemm(tb, wk, bk, nullptr, td, NL, CH, CH);                            // k
  gemm(ta, wv, bv, nullptr, te, NL, CH, CH);                            // v
  kv_kernel<<<B_N * NHD, 256, 0, stream>>>(td, te, kvb, ksb);
  la_msg<<<dim3(T_L / 16, B_N * NHD), 32, 0, stream>>>(tc, kvb, ksb, tb); // msg
  gemm(tb, w_merge, nullptr, tgt, res1, NL, CH, CH);                    // res1 = tgt + msg@Wm

  // ---- cross attention (softmax MHA, flash) ----
  ln_kernel<<<NL, CH, 0, stream>>>(res1, ln2_g, ln2_b, nullptr, ta, nullptr);
  gemm(ta, cwq, cbq, nullptr, tc, NL, CH, CH);                          // cq
  add_pos<<<4096, 256, 0, stream>>>(memory, pos_embed, bg0);            // mem
  gemm(bg0, cwk, cbk, nullptr, bg1, NS, CH, CH);                        // ck
  gemm(bg0, cwv, cbv, nullptr, bg2, NS, CH, CH);                        // cv
  flash_attn<<<dim3(T_L / 16, B_N * NHD), 32, 0, stream>>>(tc, bg1, bg2, td);
  gemm(td, cwo, cbo, res1, res2, NL, CH, CH);                           // res2

  // ---- MLP: fc1 -> dwconv -> gelu -> fc2 ----
  ln_kernel<<<NL, CH, 0, stream>>>(res2, ln3_g, ln3_b, nullptr, ta, nullptr);
  gemm(ta, mw1, mb1, nullptr, bg0, NL, CH, MLP);                        // h (8M, fits bg0..bg1)
  dwconv_gelu<<<8192, 256, 0, stream>>>(bg0, dw_k, dw_b, tb);           // h2 -> tb..te
  gemm(tb, mw2, mb2, res2, outp, NL, MLP, CH);                          // out
}